// SpiralConvCombined_77429670412668
// MI455X (gfx1250) — compile-verified
//
#include <hip/hip_runtime.h>

typedef __attribute__((ext_vector_type(2))) float v2f;
typedef __attribute__((ext_vector_type(8))) float v8f;

#define B_   16
#define C_   64
#define N_   2048
#define K_   20
#define KS_  20
#define OUT_ 128
#define NT   16
#define NTILES (N_ / NT)   // 128

// ---- dynamic LDS partition (floats) ----
#define S_OFF    0
#define S_SZ     (NT * K_ * C_)       // 20480 : gathered spirals [n][k][f]
#define ADJT_OFF (S_OFF + S_SZ)
#define ADJT_SZ  (NT * 32 * K_)       // 10240 : adj transposed [n][ks(32,zero-pad)][i(20)]
#define MT_OFF   (ADJT_OFF + ADJT_SZ)
#define MT_SZ    (NT * C_ * KS_)      // 20480 : M row-major per n [n][c=f*20+ks]
#define XT_OFF   (MT_OFF + MT_SZ)
#define XT_SZ    (NT * C_)            // 1024  : x tile [n][f]
#define W4A_OFF  (XT_OFF + XT_SZ)
#define W4A_SZ   (C_ * C_)            // 4096
#define W4D_OFF  (W4A_OFF + W4A_SZ)
#define W4D_SZ   (C_ * C_)
#define SMEM_FLOATS (W4D_OFF + W4D_SZ)   // 60416 floats = 241664 bytes

__device__ __forceinline__ v8f wmma_f32(v2f a, v2f b, v8f c) {
    // D(16x16,f32) = A(16x4,f32) * B(4x16,f32) + C
    return __builtin_amdgcn_wmma_f32_16x16x4_f32(
        false, a, false, b, (short)0, c, false, false);
}

// ------------------------------------------------------------------
// Kernel 0: transpose x[b][f][n] -> xt[b][n][f] (contiguous gather rows)
// ------------------------------------------------------------------
__global__ void xt_kernel(const float* __restrict__ x, float* __restrict__ xt) {
    __shared__ float tile[C_][C_ + 1];
    int t  = blockIdx.x;          // 512 tiles = 16 b * 32 n-chunks
    int b  = t >> 5;
    int n0 = (t & 31) * 64;
    for (int j = threadIdx.x; j < 64 * 64; j += blockDim.x) {
        int n = j & 63, f = j >> 6;                      // coalesced read over n
        tile[f][n] = x[(b * C_ + f) * N_ + n0 + n];
    }
    __syncthreads();
    for (int j = threadIdx.x; j < 64 * 64; j += blockDim.x) {
        int f = j & 63, n = j >> 6;                      // coalesced write over f
        xt[((long)(b * N_ + n0 + n)) * C_ + f] = tile[f][n];
    }
}

// ------------------------------------------------------------------
// Kernel 1: fused spiral-conv. One workgroup (4 waves) per (b, 16 n's).
// ------------------------------------------------------------------
__global__ void __launch_bounds__(128, 1)
spiral_fused(const float* __restrict__ x, const float* __restrict__ xt,
             const int* __restrict__ sidx, const float* __restrict__ adjw,
             const float* __restrict__ W2, const float* __restrict__ b2,
             const float* __restrict__ W4, float* __restrict__ pre) {
    extern __shared__ float sm[];
    float* S    = sm + S_OFF;
    float* ADJT = sm + ADJT_OFF;
    float* MT   = sm + MT_OFF;
    float* XT   = sm + XT_OFF;
    float* W4A  = sm + W4A_OFF;
    float* W4D  = sm + W4D_OFF;

    const int tid  = threadIdx.x;
    const int wave = tid >> 5;
    const int lane = tid & 31;
    const int l15  = lane & 15;
    const int hi   = lane >> 4;              // 0: lanes 0-15, 1: lanes 16-31
    const int hi2  = 2 * hi;                 // K-slice base offset for this half
    const int tileId = blockIdx.x;
    const int b  = tileId / NTILES;
    const int n0 = (tileId % NTILES) * NT;

    // ---- stage x tile as [n][f] (coalesced 16-wide over n) ----
    for (int j = tid; j < C_ * NT; j += 128) {
        int n = j & (NT - 1), f = j >> 4;
        XT[n * C_ + f] = x[(b * C_ + f) * N_ + n0 + n];
    }
    // ---- stage W4a and W4d = W4b - W4a ----
    for (int j = tid; j < C_ * C_; j += 128) {
        int f = j & 63, o = j >> 6;
        float a  = W4[o * (2 * C_) + f];
        float bb = W4[o * (2 * C_) + C_ + f];
        W4A[o * C_ + f] = a;
        W4D[o * C_ + f] = bb - a;
    }
    // ---- gather S rows (b64 per lane) + ADJ rows transposed ----
    for (int r = wave; r < NT * K_; r += 4) {
        int n = r / K_, k = r % K_;
        int idx = sidx[(b * N_ + n0 + n) * K_ + k];
        const float* src = xt + (long)idx * C_;
        *(v2f*)&S[(n * K_ + k) * C_ + 2 * lane] = *(const v2f*)(src + 2 * lane);
        // ADJT[n][ks][i] = adjw[b][n][i=k][ks]; ks rows 20..31 zeroed
        float av = (lane < KS_)
            ? adjw[(((long)(b * N_ + n0 + n)) * K_ + k) * KS_ + lane] : 0.0f;
        ADJT[(n * 32 + lane) * K_ + k] = av;
    }
    __syncthreads();

    // ============ Phase M: wave w owns f-tile w ============
    // M_n(64x20) = S_n^T(64x20) @ A_n(20x20), contraction i
    {
        const int fA = wave * 16 + l15;          // A-operand row (f)
        for (int n = 0; n < NT; ++n) {
            const float* SnA = S + n * K_ * C_ + fA;        // A: SnA[i*C_]
            const float* AnT = ADJT + n * 32 * K_;          // B: AnT[ks*K_ + i]
            for (int kst = 0; kst < 2; ++kst) {
                const int ks = kst * 16 + l15;
                const float* bp = AnT + ks * K_ + hi2;
                v8f acc = {};
#pragma unroll
                for (int i0 = 0; i0 < K_; i0 += 4) {
                    v2f a;
                    a.x = SnA[(i0 + hi2) * C_];
                    a.y = SnA[(i0 + hi2 + 1) * C_];
                    v2f bv = *(const v2f*)(bp + i0);        // ds_load_b64
                    acc = wmma_f32(a, bv, acc);
                }
                if (ks < KS_) {
                    float* mrow = MT + (wave * 16 + 8 * hi) * KS_ + ks;
#pragma unroll
                    for (int r = 0; r < 8; ++r)
                        mrow[n * (C_ * KS_) + r * KS_] = acc[r];
                }
            }
        }
    }
    __syncthreads();

    // ============ Output phases: wave w owns o-tile w ============
    const int o0 = wave * 16;
    const int oA = o0 + l15;

    // hoist W4 A-operands (k-invariant), b64 loads
    v2f a4a[16], a4d[16];
#pragma unroll
    for (int s = 0; s < 16; ++s) {
        int fb = s * 4 + hi2;
        a4a[s] = *(const v2f*)&W4A[oA * C_ + fb];
        a4d[s] = *(const v2f*)&W4D[oA * C_ + fb];
    }

    // u = (W4b - W4a) @ X   (16o x 16n), B from XT[n][f] b64
    v8f accu = {};
    {
        const float* Xb = XT + l15 * C_ + hi2;
#pragma unroll
        for (int s = 0; s < 16; ++s) {
            v2f bv = *(const v2f*)(Xb + s * 4);
            accu = wmma_f32(a4d[s], bv, accu);
        }
    }

    // v_k = W4a @ S_n[k,:]^T ; running max over k; B from S[n][k][f] b64
    v8f vmax;
#pragma unroll
    for (int r = 0; r < 8; ++r) vmax[r] = -3.402823466e38f;
    for (int k = 0; k < K_; ++k) {
        const float* Sb = S + (l15 * K_ + k) * C_ + hi2;
        v8f accv = {};
#pragma unroll
        for (int s = 0; s < 16; ++s) {
            v2f bv = *(const v2f*)(Sb + s * 4);
            accv = wmma_f32(a4a[s], bv, accv);
        }
#pragma unroll
        for (int r = 0; r < 8; ++r) vmax[r] = fmaxf(vmax[r], accv[r]);
    }

    // out2 = W2 @ M  (K=1280): A = global b64 (L2-resident), B = MT[n][c] b64
    v8f acc2 = {};
    {
        const float* W2row = W2 + (long)oA * (C_ * KS_) + hi2;
        const float* Mrow  = MT + l15 * (C_ * KS_) + hi2;
#pragma unroll 16
        for (int c0 = 0; c0 < C_ * KS_; c0 += 4) {
            if ((c0 & 63) == 0)                       // folds at compile time
                __builtin_prefetch(W2row + c0 + 128, 0, 0);
            v2f a  = *(const v2f*)(W2row + c0);       // global_load_b64
            v2f bv = *(const v2f*)(Mrow + c0);        // ds_load_b64
            acc2 = wmma_f32(a, bv, acc2);
        }
    }

    // ---- write pre-activation (out2 -> ch o, out4 -> ch 64+o) ----
#pragma unroll
    for (int r = 0; r < 8; ++r) {
        int o = o0 + r + 8 * hi;
        int n = n0 + l15;
        pre[((long)(b * OUT_ + o)) * N_ + n]      = acc2[r] + b2[o];
        pre[((long)(b * OUT_ + C_ + o)) * N_ + n] = accu[r] + vmax[r];
    }
}

// ------------------------------------------------------------------
// Kernel 2: per-channel mean / rstd (deterministic tree reduction)
// ------------------------------------------------------------------
__global__ void stats_kernel(const float* __restrict__ pre, float* __restrict__ mv) {
    __shared__ double sbuf[256], qbuf[256];
    int c = blockIdx.x;
    double s = 0.0, q = 0.0;
    for (int j = threadIdx.x; j < B_ * N_; j += 256) {
        int b = j >> 11, n = j & (N_ - 1);
        float v = pre[((long)(b * OUT_ + c)) * N_ + n];
        s += (double)v;
        q += (double)v * (double)v;
    }
    sbuf[threadIdx.x] = s; qbuf[threadIdx.x] = q;
    __syncthreads();
    for (int st = 128; st > 0; st >>= 1) {
        if (threadIdx.x < st) {
            sbuf[threadIdx.x] += sbuf[threadIdx.x + st];
            qbuf[threadIdx.x] += qbuf[threadIdx.x + st];
        }
        __syncthreads();
    }
    if (threadIdx.x == 0) {
        double inv = 1.0 / (double)(B_ * N_);
        double mean = sbuf[0] * inv;
        double var  = qbuf[0] * inv - mean * mean;
        mv[c]        = (float)mean;
        mv[OUT_ + c] = rsqrtf((float)var + 1e-5f);
    }
}

// ------------------------------------------------------------------
// Kernel 3: in-place batchnorm affine
// ------------------------------------------------------------------
__global__ void bn_kernel(float* __restrict__ out, const float* __restrict__ mv,
                          const float* __restrict__ gamma, const float* __restrict__ beta) {
    long i = (long)blockIdx.x * 256 + threadIdx.x;
    if (i >= (long)B_ * OUT_ * N_) return;
    int ch = (int)((i >> 11) & (OUT_ - 1));
    out[i] = (out[i] - mv[ch]) * mv[OUT_ + ch] * gamma[ch] + beta[ch];
}

extern "C" void kernel_launch(void* const* d_in, const int* in_sizes, int n_in,
                              void* d_out, int out_size, void* d_ws, size_t ws_size,
                              hipStream_t stream) {
    const float* x     = (const float*)d_in[0];
    const int*   sidx  = (const int*)d_in[1];
    const float* adjw  = (const float*)d_in[2];
    const float* W2    = (const float*)d_in[3];
    const float* b2    = (const float*)d_in[4];
    const float* W4    = (const float*)d_in[5];
    const float* gamma = (const float*)d_in[6];
    const float* beta  = (const float*)d_in[7];
    float* out = (float*)d_out;

    float* xt = (float*)d_ws;                       // B*N*C floats (8 MB)
    float* mv = xt + (long)B_ * N_ * C_;            // 256 floats

    size_t smem = (size_t)SMEM_FLOATS * sizeof(float);   // ~236 KB (<= 320 KB/WGP)
    hipFuncSetAttribute(reinterpret_cast<const void*>(spiral_fused),
                        hipFuncAttributeMaxDynamicSharedMemorySize, (int)smem);

    xt_kernel<<<512, 256, 0, stream>>>(x, xt);
    spiral_fused<<<B_ * NTILES, 128, smem, stream>>>(x, xt, sidx, adjw, W2, b2, W4, out);
    stats_kernel<<<OUT_, 256, 0, stream>>>(out, mv);
    long total = (long)B_ * OUT_ * N_;
    bn_kernel<<<(int)((total + 255) / 256), 256, 0, stream>>>(out, mv, gamma, beta);
}